// MultiHeadSpectralAttention_61125974556922
// MI455X (gfx1250) — compile-verified
//
#include <hip/hip_runtime.h>
#include <math.h>

// ---------------------------------------------------------------------------
// MultiHeadSpectralAttention for MI455X (gfx1250, wave32).
// fp32 end-to-end. WMMA (v_wmma_f32_16x16x4_f32) for the gate MLP GEMMs
// (branch-free inner loop: A staged zero-padded in LDS, B column-clamped);
// LDS radix-2 FFT with LDS twiddle table for the length-2048 global path;
// per-thread 16-pt DFT + global_atomic_add_f32 overlap-add for the local path.
// ---------------------------------------------------------------------------

typedef __attribute__((ext_vector_type(2))) float v2f;
typedef __attribute__((ext_vector_type(8))) float v8f;

#define B_    8
#define N_    2048
#define D_    768
#define H_    12
#define HD_   64
#define W_    16
#define HOP_  8
#define GB_   1025
#define LB_   9
#define NW_   254
#define OUTG_ (H_ * GB_)   // 12300
#define OUTL_ (H_ * LB_)   // 108
#define LOG2N 11
#define INV_SQRT_N 0.022097086912079608f   // 1/sqrt(2048)
#define TWO_PI 6.283185307179586f
#define EPS_LN 1e-5f

// ---------------------------------------------------------------------------
// Kernel 1: ctx = mean over N of x  -> (B, D)
// ---------------------------------------------------------------------------
__global__ void ctx_mean_kernel(const float* __restrict__ x,
                                float* __restrict__ ctx) {
    int b     = blockIdx.x / (D_ / 256);
    int chunk = blockIdx.x % (D_ / 256);
    int d     = chunk * 256 + threadIdx.x;
    const float* p = x + (size_t)b * N_ * D_ + d;
    float s = 0.0f;
    for (int n = 0; n < N_; ++n) {
        __builtin_prefetch(p + 24 * D_, 0, 0);  // global_prefetch_b8
        s += *p;
        p += D_;
    }
    ctx[b * D_ + d] = s * (1.0f / (float)N_);
}

// ---------------------------------------------------------------------------
// Kernel 2: gate GEMM with fp32 WMMA (branch-free inner loop).
//   out[r, c] = act( sum_k A[r,k] * Wt[k,c] + bias[c] ),  r in [0,8)
// Block = 256 threads = 8 waves; block covers 128 output columns.
// A (8xK) staged into LDS zero-padded to 16xK; each wave owns one 16-col tile.
// A-frag layout (ISA 16x4 f32): lanes 0-15 -> K=k0,k0+1 ; lanes 16-31 -> k0+2,k0+3.
// C/D layout: VGPR i, lanes 0-15 -> row i (rows 0..7), lanes 16-31 -> row i+8.
// ---------------------------------------------------------------------------
__device__ __forceinline__ float gelu_exact(float v) {
    return 0.5f * v * (1.0f + erff(v * 0.7071067811865475f));
}

template <int ACT, int K>  // ACT: 0 = gelu, 1 = tanh
__global__ void gate_gemm_wmma(const float* __restrict__ A,    // 8 x K (row-major)
                               const float* __restrict__ Wt,   // K x OUT (row-major)
                               const float* __restrict__ bias, // OUT
                               float* __restrict__ out,        // 8 x OUT
                               int OUT) {
    __shared__ float As[16 * K];                 // zero-padded A, 48 KB for K=768

    const int tid = threadIdx.x;
    for (int idx = tid; idx < 16 * K; idx += 256) {
        As[idx] = (idx < 8 * K) ? A[idx] : 0.0f;
    }
    __syncthreads();

    const int wave = tid >> 5;                   // 0..7
    const int lane = tid & 31;
    const int half = lane >> 4;                  // 0 | 1
    const int l15  = lane & 15;
    const int c    = blockIdx.x * 128 + wave * 16 + l15;
    const int cc   = (c < OUT) ? c : (OUT - 1);  // clamp: loads stay legal

    v8f acc = {};
    #pragma unroll 8
    for (int k0 = 0; k0 < K; k0 += 4) {
        const int ka = k0 + 2 * half;
        const v2f a = *(const v2f*)&As[l15 * K + ka];   // ds_load_b64
        v2f bf;
        bf.x = Wt[(size_t)ka * OUT + cc];
        bf.y = Wt[(size_t)(ka + 1) * OUT + cc];
        acc = __builtin_amdgcn_wmma_f32_16x16x4_f32(
                  false, a, false, bf, (short)0, acc, false, false);
    }

    if (half == 0 && c < OUT) {                  // rows 0..7 live in lanes 0..15
        const float bv = bias[c];
        #pragma unroll
        for (int r = 0; r < 8; ++r) {
            float v = acc[r] + bv;
            v = (ACT == 0) ? gelu_exact(v) : tanhf(v);
            out[(size_t)r * OUT + c] = v;
        }
    }
}

// ---------------------------------------------------------------------------
// Kernel 3: global spectral filter. One workgroup (128 threads) per
// (b,h,hd) vector: LDS radix-2 DIT FFT (2048 pts) with LDS twiddle table,
// ortho gain+modrelu on bins 0..1024, Hermitian fill, bit-reverse,
// inverse DIT, write fused = fusion_w[0] * x_global.
// ---------------------------------------------------------------------------
__device__ __forceinline__ int bitrev11(int i) {
    return (int)(__brev((unsigned)i) >> (32 - LOG2N));
}

__global__ void global_spectral_kernel(const float* __restrict__ x,
                                       const float* __restrict__ baseg,  // H*GB
                                       const float* __restrict__ mrbg,   // H*GB
                                       const float* __restrict__ mg,     // B x (H*GB)
                                       const float* __restrict__ fusion_w,
                                       float* __restrict__ fused) {      // B,H,N,HD
    __shared__ float re[N_];
    __shared__ float im[N_];
    __shared__ float twc[N_ / 2];
    __shared__ float tws[N_ / 2];

    const int gid = blockIdx.x;            // b*H*HD + h*HD + hd
    const int hd  = gid % HD_;
    const int h   = (gid / HD_) % H_;
    const int b   = gid / (HD_ * H_);
    const int tid = threadIdx.x;           // 0..127

    // --- twiddle table: tw[t] = exp(-2*pi*i*t/N), t in [0, N/2) ---
    for (int t = tid; t < N_ / 2; t += 128) {
        float sn, cs;
        __sincosf((-TWO_PI / (float)N_) * (float)t, &sn, &cs);
        twc[t] = cs; tws[t] = sn;
    }

    // --- load with bit-reversed placement ---
    const float* xp = x + (size_t)b * N_ * D_ + h * HD_ + hd;
    for (int i = tid; i < N_; i += 128) {
        const int r = bitrev11(i);
        re[r] = xp[(size_t)i * D_];
        im[r] = 0.0f;
    }
    __syncthreads();

    // --- forward FFT (sign -1) ---
    for (int s = 1; s <= LOG2N; ++s) {
        const int hl = 1 << (s - 1);
        const int tshift = LOG2N - s;
        for (int j = tid; j < N_ / 2; j += 128) {
            const int grp = j >> (s - 1), pos = j & (hl - 1);
            const int i0 = (grp << s) + pos, i1 = i0 + hl;
            const int ti = pos << tshift;
            const float cs = twc[ti], sn = tws[ti];
            const float tr = cs * re[i1] - sn * im[i1];
            const float tim = cs * im[i1] + sn * re[i1];
            re[i1] = re[i0] - tr;  im[i1] = im[i0] - tim;
            re[i0] += tr;          im[i0] += tim;
        }
        __syncthreads();
    }

    // --- spectral gain + modrelu on ortho-normalized bins 0..1024 ---
    float zr[9], zi[9];
    int nb = 0;
    const float fw0 = fusion_w[0];
    for (int k = tid; k <= N_ / 2; k += 128) {
        float fr = re[k] * INV_SQRT_N;
        float fi = im[k] * INV_SQRT_N;
        const float gain = baseg[h * GB_ + k] + mg[(size_t)b * OUTG_ + h * GB_ + k];
        fr *= gain; fi *= gain;
        const float a   = sqrtf(fr * fr + fi * fi);
        const float fac = fmaxf(a + mrbg[h * GB_ + k], 0.0f) / fmaxf(a, 1e-6f);
        zr[nb] = fr * fac; zi[nb] = fi * fac; ++nb;
    }
    __syncthreads();

    // --- write filtered spectrum + Hermitian mirror ---
    nb = 0;
    for (int k = tid; k <= N_ / 2; k += 128) {
        re[k] = zr[nb]; im[k] = zi[nb];
        if (k >= 1 && k < N_ / 2) { re[N_ - k] = zr[nb]; im[N_ - k] = -zi[nb]; }
        ++nb;
    }
    __syncthreads();

    // --- bit-reverse permute (disjoint pairs) ---
    for (int i = tid; i < N_; i += 128) {
        const int j = bitrev11(i);
        if (i < j) {
            float t = re[i]; re[i] = re[j]; re[j] = t;
            t = im[i]; im[i] = im[j]; im[j] = t;
        }
    }
    __syncthreads();

    // --- inverse FFT (sign +1): conjugate twiddles ---
    for (int s = 1; s <= LOG2N; ++s) {
        const int hl = 1 << (s - 1);
        const int tshift = LOG2N - s;
        for (int j = tid; j < N_ / 2; j += 128) {
            const int grp = j >> (s - 1), pos = j & (hl - 1);
            const int i0 = (grp << s) + pos, i1 = i0 + hl;
            const int ti = pos << tshift;
            const float cs = twc[ti], sn = -tws[ti];
            const float tr = cs * re[i1] - sn * im[i1];
            const float tim = cs * im[i1] + sn * re[i1];
            re[i1] = re[i0] - tr;  im[i1] = im[i0] - tim;
            re[i0] += tr;          im[i0] += tim;
        }
        __syncthreads();
    }

    // --- fused init = fw0 * irfft(ortho) ---
    float* fp = fused + ((size_t)(b * H_ + h) * N_) * HD_ + hd;
    for (int n = tid; n < N_; n += 128) {
        fp[(size_t)n * HD_] = fw0 * INV_SQRT_N * re[n];
    }
}

// ---------------------------------------------------------------------------
// Kernel 4: local STFT path. One thread per (b,h,win,hd). Constant trig
// tables (cos/sin of 2*pi*j/16), Hann window, 16-pt rDFT, gain + modrelu,
// 16-pt inverse, Hann again, atomic overlap-add into fused.
// ---------------------------------------------------------------------------
__global__ void local_spectral_kernel(const float* __restrict__ x,
                                      const float* __restrict__ basel,  // H*LB
                                      const float* __restrict__ mrbl,   // H*LB
                                      const float* __restrict__ ml,     // B x (H*LB)
                                      const float* __restrict__ fusion_w,
                                      float* __restrict__ fused) {
    const float c16[16] = {
         1.0f,  0.92387953251f,  0.70710678119f,  0.38268343236f,
         0.0f, -0.38268343236f, -0.70710678119f, -0.92387953251f,
        -1.0f, -0.92387953251f, -0.70710678119f, -0.38268343236f,
         0.0f,  0.38268343236f,  0.70710678119f,  0.92387953251f };
    const float s16[16] = {
         0.0f,  0.38268343236f,  0.70710678119f,  0.92387953251f,
         1.0f,  0.92387953251f,  0.70710678119f,  0.38268343236f,
         0.0f, -0.38268343236f, -0.70710678119f, -0.92387953251f,
        -1.0f, -0.92387953251f, -0.70710678119f, -0.38268343236f };

    const long t = (long)blockIdx.x * blockDim.x + threadIdx.x;
    const int hd = (int)(t % HD_);
    long r = t / HD_;
    const int win = (int)(r % NW_); r /= NW_;
    const int h   = (int)(r % H_);
    const int b   = (int)(r / H_);
    if (b >= B_) return;

    float hw[16], v[16];
    const float* xp = x + ((size_t)b * N_ + win * HOP_) * D_ + h * HD_ + hd;
    #pragma unroll
    for (int j = 0; j < 16; ++j) {
        hw[j] = 0.5f * (1.0f - c16[j]);          // Hann
        v[j]  = xp[(size_t)j * D_] * hw[j];
    }

    float Zr[9], Zi[9];
    #pragma unroll
    for (int k = 0; k <= 8; ++k) {
        float sr = 0.0f, si = 0.0f;
        #pragma unroll
        for (int j = 0; j < 16; ++j) {
            const int a = (k * j) & 15;
            sr += v[j] * c16[a];
            si -= v[j] * s16[a];
        }
        sr *= 0.25f; si *= 0.25f;                // ortho: 1/sqrt(16)
        const float gain = basel[h * LB_ + k] + ml[(size_t)b * OUTL_ + h * LB_ + k];
        sr *= gain; si *= gain;
        const float a2  = sqrtf(sr * sr + si * si);
        const float fac = fmaxf(a2 + mrbl[h * LB_ + k], 0.0f) / fmaxf(a2, 1e-6f);
        Zr[k] = sr * fac; Zi[k] = si * fac;
    }

    const float fw1 = fusion_w[1];
    float* fp = fused + ((size_t)(b * H_ + h) * N_ + win * HOP_) * HD_ + hd;
    #pragma unroll
    for (int j = 0; j < 16; ++j) {
        float y = Zr[0] + Zr[8] * c16[(8 * j) & 15];
        #pragma unroll
        for (int k = 1; k < 8; ++k) {
            const int a = (k * j) & 15;
            y += 2.0f * (Zr[k] * c16[a] - Zi[k] * s16[a]);
        }
        y *= 0.25f;                              // ortho inverse
        atomicAdd(&fp[(size_t)j * HD_], fw1 * y * hw[j]);
    }
}

// ---------------------------------------------------------------------------
// Kernel 5: residual + LayerNorm over D. One block per (b,n).
// ---------------------------------------------------------------------------
__global__ void residual_ln_kernel(const float* __restrict__ x,
                                   const float* __restrict__ fused,
                                   const float* __restrict__ gamma,
                                   const float* __restrict__ beta,
                                   float* __restrict__ out) {
    __shared__ float sbuf[256];
    __shared__ float s2buf[256];
    const int n = blockIdx.x % N_;
    const int b = blockIdx.x / N_;
    const int tid = threadIdx.x;

    float vals[D_ / 256];
    float s = 0.0f, s2 = 0.0f;
    #pragma unroll
    for (int i = 0; i < D_ / 256; ++i) {
        const int d  = i * 256 + tid;
        const int h  = d >> 6, hd = d & 63;
        const float y = x[((size_t)b * N_ + n) * D_ + d] +
                        fused[((size_t)(b * H_ + h) * N_ + n) * HD_ + hd];
        vals[i] = y; s += y; s2 += y * y;
    }
    sbuf[tid] = s; s2buf[tid] = s2;
    __syncthreads();
    for (int o = 128; o > 0; o >>= 1) {
        if (tid < o) { sbuf[tid] += sbuf[tid + o]; s2buf[tid] += s2buf[tid + o]; }
        __syncthreads();
    }
    const float mu   = sbuf[0] * (1.0f / (float)D_);
    const float var  = s2buf[0] * (1.0f / (float)D_) - mu * mu;
    const float rstd = rsqrtf(var + EPS_LN);
    #pragma unroll
    for (int i = 0; i < D_ / 256; ++i) {
        const int d = i * 256 + tid;
        out[((size_t)b * N_ + n) * D_ + d] = (vals[i] - mu) * rstd * gamma[d] + beta[d];
    }
}

// ---------------------------------------------------------------------------
// Launch
// ---------------------------------------------------------------------------
extern "C" void kernel_launch(void* const* d_in, const int* in_sizes, int n_in,
                              void* d_out, int out_size, void* d_ws, size_t ws_size,
                              hipStream_t stream) {
    const float* x      = (const float*)d_in[0];
    const float* baseg  = (const float*)d_in[1];
    const float* basel  = (const float*)d_in[2];
    const float* mrbg   = (const float*)d_in[3];
    const float* mrbl   = (const float*)d_in[4];
    const float* w1g    = (const float*)d_in[5];
    const float* b1g    = (const float*)d_in[6];
    const float* w2g    = (const float*)d_in[7];
    const float* b2g    = (const float*)d_in[8];
    const float* w1l    = (const float*)d_in[9];
    const float* b1l    = (const float*)d_in[10];
    const float* w2l    = (const float*)d_in[11];
    const float* b2l    = (const float*)d_in[12];
    const float* fusion = (const float*)d_in[13];
    const float* gamma  = (const float*)d_in[14];
    const float* beta   = (const float*)d_in[15];
    float* out = (float*)d_out;

    // workspace layout (floats)
    float* ws    = (float*)d_ws;
    float* ctx   = ws;                       // 8*768
    float* hg    = ctx + B_ * D_;            // 8*768
    float* hl    = hg  + B_ * D_;            // 8*768
    float* mg    = hl  + B_ * D_;            // 8*12300
    float* ml    = mg  + B_ * OUTG_;         // 8*108
    float* fused = ml  + B_ * OUTL_;         // B*H*N*HD (~50 MB)

    // 1. context mean
    ctx_mean_kernel<<<B_ * (D_ / 256), 256, 0, stream>>>(x, ctx);

    // 2. gate MLPs (WMMA fp32); each block covers 128 output columns
    gate_gemm_wmma<0, D_><<<(D_ + 127) / 128, 256, 0, stream>>>(ctx, w1g, b1g, hg, D_);
    gate_gemm_wmma<1, D_><<<(OUTG_ + 127) / 128, 256, 0, stream>>>(hg, w2g, b2g, mg, OUTG_);
    gate_gemm_wmma<0, D_><<<(D_ + 127) / 128, 256, 0, stream>>>(ctx, w1l, b1l, hl, D_);
    gate_gemm_wmma<1, D_><<<(OUTL_ + 127) / 128, 256, 0, stream>>>(hl, w2l, b2l, ml, OUTL_);

    // 3. global spectral path (also initializes `fused`)
    global_spectral_kernel<<<B_ * H_ * HD_, 128, 0, stream>>>(
        x, baseg, mrbg, mg, fusion, fused);

    // 4. local STFT path (atomic overlap-add into `fused`)
    {
        const long total = (long)B_ * H_ * NW_ * HD_;
        const int blocks = (int)((total + 255) / 256);
        local_spectral_kernel<<<blocks, 256, 0, stream>>>(
            x, basel, mrbl, ml, fusion, fused);
    }

    // 5. residual + LayerNorm
    residual_ln_kernel<<<B_ * N_, 256, 0, stream>>>(x, fused, gamma, beta, out);
}